// DCT_Forward_Model_27616639714073
// MI455X (gfx1250) — compile-verified
//
#include <hip/hip_runtime.h>

// CDNA5 / gfx1250: wave32, WMMA f32 16x16x4.
typedef __attribute__((ext_vector_type(2))) float v2f;   // A/B operand: 16x4 or 4x16 f32 = 2 VGPRs
typedef __attribute__((ext_vector_type(8))) float v8f;   // C/D: 16x16 f32 = 8 VGPRs

// JPEG Q50 luminance table (row-major), matches reference.
__device__ __constant__ float kQ50[64] = {
  16.f,11.f,10.f,16.f,24.f,40.f,51.f,61.f,
  12.f,12.f,14.f,19.f,26.f,58.f,60.f,55.f,
  14.f,13.f,16.f,24.f,40.f,57.f,69.f,56.f,
  14.f,17.f,22.f,29.f,51.f,87.f,80.f,62.f,
  18.f,22.f,37.f,56.f,68.f,109.f,103.f,77.f,
  24.f,35.f,55.f,64.f,81.f,104.f,113.f,92.f,
  49.f,64.f,78.f,87.f,103.f,121.f,120.f,101.f,
  72.f,92.f,95.f,98.f,112.f,100.f,103.f,99.f
};

// Full K=8 product: two chained V_WMMA_F32_16X16X4_F32 (K chunks 0..3 and 4..7).
static __device__ __forceinline__ v8f wmma_k8(v2f a0, v2f a1, v2f b0, v2f b1) {
  v8f c = {};
  c = __builtin_amdgcn_wmma_f32_16x16x4_f32(false, a0, false, b0, (short)0, c, false, false);
  c = __builtin_amdgcn_wmma_f32_16x16x4_f32(false, a1, false, b1, (short)0, c, false, false);
  return c;
}

__global__ __launch_bounds__(256) void dct_q50_kernel(
    const float* __restrict__ in,   // [S,1,32,32]
    const float* __restrict__ wgt,  // [1,64] DCT matrix row-major
    const float* __restrict__ wfac, // [1,1] scale factor (65000)
    float* __restrict__ out,        // [1,1,S,32,32]
    int npairs)                     // total 8x8 tiles / 2
{
  __shared__ __align__(16) float smem[8 * 128];  // one 2-tile slot per wave (8 waves)
  const int tid   = threadIdx.x;
  const int lane  = tid & 31;
  const int wslot = tid >> 5;
  float* S = &smem[wslot * 128];

  const float sf  = wfac[0];
  const float rsf = 1.0f / sf;

  // Operand index helpers (ISA §7.12.2 layouts):
  const int r   = lane & 7;          // A-matrix row (rows 8..15 duplicate 0..7)
  const int n   = lane & 7;          // B/D column within a block
  const int h   = lane >> 4;         // lane half -> K upper-half offset
  const int blk = (lane >> 3) & 1;   // which of the 2 blocks this lane's column maps to

  // A operands: Xs = weight*sf (stages 1,2), Xsᵀ (stages 3,4). A(m,k): lane m + 16*(k>>1), vgpr k&1.
  v2f aX0, aX1, aT0, aT1;
#pragma unroll
  for (int v = 0; v < 2; ++v) {
    const int cl = v + 2 * h;                 // K within 4-chunk
    aX0[v] = wgt[r * 8 + cl]       * sf;      // Xs[r][cl]
    aX1[v] = wgt[r * 8 + cl + 4]   * sf;      // Xs[r][cl+4]
    aT0[v] = wgt[cl * 8 + r]       * sf;      // Xsᵀ[r][cl] = Xs[cl][r]
    aT1[v] = wgt[(cl + 4) * 8 + r] * sf;      // Xsᵀ[r][cl+4]
  }

  // Per-lane quant row: Q50[n][j] and its reciprocal (hoists IEEE div out of loop).
  float qrow[8], rq[8];
#pragma unroll
  for (int j = 0; j < 8; ++j) {
    qrow[j] = kQ50[n * 8 + j];
    rq[j]   = 1.0f / qrow[j];
  }

  // Wave-uniform loop bounds, forced into SGPRs so address math stays scalar.
  const int nwaves = gridDim.x * (blockDim.x >> 5);
  const int wid    = __builtin_amdgcn_readfirstlane(
                       blockIdx.x * (blockDim.x >> 5) + wslot);

  // Loop-invariant per-lane byte offsets (element units here).
  const int lblk = lane >> 4;                       // cooperative-load indices
  const int lrow = (lane >> 1) & 7;
  const int lc0  = (lane & 1) * 4;
  const int goff_in  = lblk * 8 + lrow * 32 + lc0;  // + f(p) -> input float4 address
  const int goff_out = blk * 8 + n * 32;            // + f(p) -> output row address (lanes<16)
  const int soff_ld  = lblk * 64 + lrow * 8 + lc0;  // LDS cooperative store
  const int sbaseN0  = blk * 64 + (2 * h) * 8 + n;  // mode-N read, kc=0 (K=2h)
  const int sbaseT   = blk * 64 + n * 8 + 2 * h;    // mode-T read base
  const int sstoreA  = blk * 64 + n;                // full 8-row store base

  for (int p = wid; p < npairs; p += nwaves) {
    // f(p) = tile_base(2p): scalar (p is wave-uniform).
    const int fp = ((p >> 3) << 10) + (((p >> 1) & 3) << 8) + ((p & 1) << 4);

    // ---- load two 8x8 tiles into LDS row-major, centered (-128), coalesced b128
    {
      const float4 dv = *(const float4*)(in + fp + goff_in);
      float4 sv;
      sv.x = dv.x - 128.f; sv.y = dv.y - 128.f; sv.z = dv.z - 128.f; sv.w = dv.w - 128.f;
      *(float4*)(S + soff_ld) = sv;
    }
    asm volatile("s_wait_dscnt 0x0" ::: "memory");

    v2f b0, b1;

    // ---- stage 1: t11 = (Xs . B)/sf   (B operand untransposed: B(K,n)=blk[K][n])
#pragma unroll
    for (int v = 0; v < 2; ++v) {
      b0[v] = S[sbaseN0 + v * 8];
      b1[v] = S[sbaseN0 + (v + 4) * 8];
    }
    v8f c = wmma_k8(aX0, aX1, b0, b1);
#pragma unroll
    for (int i = 0; i < 8; ++i) c[i] *= rsf;
#pragma unroll
    for (int i = 0; i < 8; ++i) S[sstoreA + i * 8] = c[i];  // halves write identical values
    asm volatile("s_wait_dscnt 0x0" ::: "memory");

    // ---- stage 2: t22ᵀ = (Xs . t11ᵀ)/sf   (transposed read), then quant/dequant
#pragma unroll
    for (int v = 0; v < 2; ++v) {
      b0[v] = S[sbaseT + v];
      b1[v] = S[sbaseT + v + 4];
    }
    c = wmma_k8(aX0, aX1, b0, b1);
#pragma unroll
    for (int i = 0; i < 8; ++i) {
      const float w = c[i] * rsf;                   // t22(n, i)
      const float q = rintf(w * rq[i] + 1e-6f);     // jnp.round = round-half-even
      c[i] = qrow[i] * q;                           // deq(n, i)
    }
#pragma unroll
    for (int i = 0; i < 8; ++i) S[sstoreA + i * 8] = c[i];  // S now holds deqᵀ
    asm volatile("s_wait_dscnt 0x0" ::: "memory");

    // ---- stage 3: t = (Xsᵀ . deq)/sf   (transposed read of deqᵀ gives deq)
#pragma unroll
    for (int v = 0; v < 2; ++v) {
      b0[v] = S[sbaseT + v];
      b1[v] = S[sbaseT + v + 4];
    }
    c = wmma_k8(aT0, aT1, b0, b1);
#pragma unroll
    for (int i = 0; i < 8; ++i) c[i] *= rsf;
#pragma unroll
    for (int i = 0; i < 8; ++i) S[sstoreA + i * 8] = c[i];  // S holds t row-major
    asm volatile("s_wait_dscnt 0x0" ::: "memory");

    // ---- stage 4: t2ᵀ = (Xsᵀ . tᵀ)/sf
#pragma unroll
    for (int v = 0; v < 2; ++v) {
      b0[v] = S[sbaseT + v];
      b1[v] = S[sbaseT + v + 4];
    }
    c = wmma_k8(aT0, aT1, b0, b1);

    // ---- output: out(n, j) = rint(t2(n, j)) + 128 ; D(r=j, col=n) = t2(n, j)
    if (lane < 16) {                                // upper half duplicates lower
      float4 o0, o1;
      o0.x = rintf(c[0] * rsf) + 128.f; o0.y = rintf(c[1] * rsf) + 128.f;
      o0.z = rintf(c[2] * rsf) + 128.f; o0.w = rintf(c[3] * rsf) + 128.f;
      o1.x = rintf(c[4] * rsf) + 128.f; o1.y = rintf(c[5] * rsf) + 128.f;
      o1.z = rintf(c[6] * rsf) + 128.f; o1.w = rintf(c[7] * rsf) + 128.f;
      float* op = out + fp + goff_out;              // contiguous 32B row per lane
      *(float4*)(op)     = o0;
      *(float4*)(op + 4) = o1;
    }
  }
}

extern "C" void kernel_launch(void* const* d_in, const int* in_sizes, int n_in,
                              void* d_out, int out_size, void* d_ws, size_t ws_size,
                              hipStream_t stream) {
  const float* in  = (const float*)d_in[0];   // [S,1,32,32] f32
  const float* w   = (const float*)d_in[1];   // [1,64] f32
  const float* wf  = (const float*)d_in[2];   // [1,1] f32
  float* out = (float*)d_out;

  const int npairs = in_sizes[0] / 128;       // (#8x8 tiles)/2
  int blocks = (npairs + 7) / 8;              // 8 waves per 256-thread block, 1 pair/wave/iter
  if (blocks > 1024) blocks = 1024;
  if (blocks < 1) blocks = 1;
  dct_q50_kernel<<<blocks, 256, 0, stream>>>(in, w, wf, out, npairs);
}